// HybridModel_23244363006240
// MI455X (gfx1250) — compile-verified
//
#include <hip/hip_runtime.h>

#define NST 65536   // 2^16 amplitudes
#define NTH 1024    // threads per simulation block

typedef __attribute__((ext_vector_type(2))) float v2f;
typedef __attribute__((ext_vector_type(8))) float v8f;

// ---------------------------------------------------------------------------
// Workgroup-wide sync with WGP-scope memory visibility (all waves of a block
// run on one WGP; global traffic is WGP-cache coherent at that scope).
// ---------------------------------------------------------------------------
__device__ __forceinline__ void gsync() {
  __threadfence_block();
  __syncthreads();
}

__device__ __forceinline__ int insert_zero(int x, int p) {
  int mask = (1 << p) - 1;
  return ((x & ~mask) << 1) | (x & mask);
}

// 2x2 complex rotations on register pairs
__device__ __forceinline__ void ry2(float& r0, float& u0, float& r1, float& u1,
                                    float c, float s) {
  float a = r0, bb = u0, d = r1, e = u1;
  r0 = c * a - s * d;  u0 = c * bb - s * e;
  r1 = s * a + c * d;  u1 = s * bb + c * e;
}
__device__ __forceinline__ void rx2(float& r0, float& u0, float& r1, float& u1,
                                    float c, float s) {
  // RX: [[c, -i s], [-i s, c]]
  float a = r0, bb = u0, d = r1, e = u1;
  r0 = c * a + s * e;  u0 = c * bb - s * d;
  r1 = c * d + s * bb; u1 = c * e - s * a;
}
__device__ __forceinline__ void swap2(float& r0, float& u0, float& r1, float& u1) {
  float t = r0; r0 = r1; r1 = t;
  t = u0; u0 = u1; u1 = t;
}

// Local-group gate application on a 16-amplitude register block.
template <int LB>
__device__ __forceinline__ void ry_local(float (&r)[16], float (&u)[16], float c, float s) {
  #pragma unroll
  for (int j = 0; j < 16; ++j)
    if (!(j & (1 << LB)))
      ry2(r[j], u[j], r[j | (1 << LB)], u[j | (1 << LB)], c, s);
}
template <int LB>
__device__ __forceinline__ void rx_local(float (&r)[16], float (&u)[16], float c, float s) {
  #pragma unroll
  for (int j = 0; j < 16; ++j)
    if (!(j & (1 << LB)))
      rx2(r[j], u[j], r[j | (1 << LB)], u[j | (1 << LB)], c, s);
}
template <int LC, int LT>
__device__ __forceinline__ void cnot_local(float (&r)[16], float (&u)[16]) {
  #pragma unroll
  for (int j = 0; j < 16; ++j)
    if ((j & (1 << LC)) && !(j & (1 << LT)))
      swap2(r[j], u[j], r[j | (1 << LT)], u[j | (1 << LT)]);
}

// ---------------------------------------------------------------------------
// One full conv iteration as a single sweep over 16-amplitude groups.
// Bits touched by iteration i: pa=15-i (wire i), pb=14-i (wire i+1),
// pd=12-i (wire i+3), pe=11-i (wire i+4);  local bit0=pe,1=pd,2=pb,3=pa.
// Gate order:
//   RY(c0) wire i | CNOT(i->i+3) | RY(c1) wire i+3 | RY(c2) wire i+1 |
//   CNOT(i+1->i+4) | CNOT(i+3->i+1) | RX(c3) wire i+4 | RX(c4) wire i+1
// FIRST: source amplitudes from the encoder product-state tables (no read).
// LAST : fold |amp|^2 * sign(P(idx)) into acc[] and skip the state write,
//        where P is the final 36-CNOT cascade as a GF(2) linear map.
// ---------------------------------------------------------------------------
template <bool FIRST, bool LAST>
__device__ __forceinline__ void conv_iter(float* __restrict__ re, float* __restrict__ im,
                                          int i, const float* __restrict__ thi,
                                          const float* __restrict__ tlo,
                                          const float (&c)[5], const float (&s)[5],
                                          float (&acc)[16]) {
  const int pa = 15 - i, pb = 14 - i, pd = 12 - i, pe = 11 - i;  // pe<pd<pb<pa
  for (int m = threadIdx.x; m < (NST >> 4); m += NTH) {
    int base = insert_zero(insert_zero(insert_zero(insert_zero(m, pe), pd), pb), pa);
    int off[16];
    #pragma unroll
    for (int j = 0; j < 16; ++j) {
      int o = base;
      if (j & 1) o |= 1 << pe;
      if (j & 2) o |= 1 << pd;
      if (j & 4) o |= 1 << pb;
      if (j & 8) o |= 1 << pa;
      off[j] = o;
    }
    float r[16], u[16];
    if (FIRST) {
      #pragma unroll
      for (int j = 0; j < 16; ++j) {
        r[j] = thi[off[j] >> 8] * tlo[off[j] & 255];
        u[j] = 0.f;
      }
    } else {
      #pragma unroll
      for (int j = 0; j < 16; ++j) { r[j] = re[off[j]]; u[j] = im[off[j]]; }
    }

    ry_local<3>(r, u, c[0], s[0]);   // RY wire i
    cnot_local<3, 1>(r, u);          // CNOT i -> i+3
    ry_local<1>(r, u, c[1], s[1]);   // RY wire i+3
    ry_local<2>(r, u, c[2], s[2]);   // RY wire i+1
    cnot_local<2, 0>(r, u);          // CNOT i+1 -> i+4
    cnot_local<1, 2>(r, u);          // CNOT i+3 -> i+1
    rx_local<0>(r, u, c[3], s[3]);   // RX wire i+4
    rx_local<2>(r, u, c[4], s[4]);   // RX wire i+1

    if (LAST) {
      #pragma unroll
      for (int j = 0; j < 16; ++j) {
        float p = r[j] * r[j] + u[j] * u[j];
        int idx = off[j];
        int pj = idx ^ ((((idx >> 1) ^ (idx >> 3) ^ (idx >> 4)) & 0xFFF0));
        #pragma unroll
        for (int w = 0; w < 16; ++w)
          acc[w] = fmaf(((pj >> (15 - w)) & 1) ? -1.0f : 1.0f, p, acc[w]);
      }
    } else {
      #pragma unroll
      for (int j = 0; j < 16; ++j) { re[off[j]] = r[j]; im[off[j]] = u[j]; }
    }
  }
}

// ---------------------------------------------------------------------------
// Kernel A: AvgPool2d(7,7): (16,1,28,28) -> xd (16,16)
// ---------------------------------------------------------------------------
__global__ void avgpool_kernel(const float* __restrict__ x, float* __restrict__ xd) {
  int t = threadIdx.x;            // 0..255
  int b = t >> 4, i = t & 15;
  int r0 = (i >> 2) * 7, c0 = (i & 3) * 7;
  const float* img = x + b * 784;
  float s = 0.f;
  #pragma unroll
  for (int dr = 0; dr < 7; ++dr)
    #pragma unroll
    for (int dc = 0; dc < 7; ++dc)
      s += img[(r0 + dr) * 28 + (c0 + dc)];
  xd[t] = s * (1.0f / 49.0f);
}

// ---------------------------------------------------------------------------
// Kernel B: one block per (batch, kernel) pair -> 32 blocks.
// 16-qubit state-vector sim, L2-resident, maximally gate-fused:
//   - encoder collapsed into product-state tables, sourced in iteration 0
//   - each conv iteration = ONE 16-amplitude-group register sweep (12 total)
//   - final 36-CNOT cascade + expz reduction folded into iteration 11
// Total: 12 sweeps, 11 block barriers.
// ---------------------------------------------------------------------------
__global__ void __launch_bounds__(NTH)
qsim_kernel(const float* __restrict__ xd, const float* __restrict__ kp,
            float* __restrict__ re_all, float* __restrict__ im_all,
            float* __restrict__ feats) {
  const int blk = blockIdx.x;     // 0..31
  const int b = blk >> 1;
  const int k = blk & 1;
  float* re = re_all + (size_t)blk * NST;
  float* im = im_all + (size_t)blk * NST;
  const int tid = threadIdx.x;

  __shared__ float cw[16], sw[16];
  __shared__ float thi[256], tlo[256];
  __shared__ float red[32 * 16];

  // Per-wire encoder cos/sin
  if (tid < 16) {
    float th = 0.5f * xd[b * 16 + tid];
    cw[tid] = cosf(th);
    sw[tid] = sinf(th);
  }
  __syncthreads();

  // Product tables: idx bit (15-w) = wire w.  High byte = wires 0..7.
  if (tid < 256) {
    float p = 1.f;
    #pragma unroll
    for (int w = 0; w < 8; ++w)
      p *= ((tid >> (7 - w)) & 1) ? sw[w] : cw[w];
    thi[tid] = p;
  } else if (tid < 512) {
    int t = tid - 256;
    float p = 1.f;
    #pragma unroll
    for (int w = 8; w < 16; ++w)
      p *= ((t >> (15 - w)) & 1) ? sw[w] : cw[w];
    tlo[t] = p;
  }
  __syncthreads();

  float c[5], s[5];
  #pragma unroll
  for (int j = 0; j < 5; ++j) {
    float th = 0.5f * kp[k * 5 + j];
    c[j] = cosf(th);
    s[j] = sinf(th);
  }

  float acc[16];
  #pragma unroll
  for (int w = 0; w < 16; ++w) acc[w] = 0.f;

  // Iteration 0 sources from tables; iterations 1..10 stream the state;
  // iteration 11 sinks directly into the expz accumulators.
  conv_iter<true, false>(re, im, 0, thi, tlo, c, s, acc);
  gsync();
  for (int i = 1; i <= 10; ++i) {
    conv_iter<false, false>(re, im, i, thi, tlo, c, s, acc);
    gsync();
  }
  conv_iter<false, true>(re, im, 11, thi, tlo, c, s, acc);

  // wave32 reduction, then cross-wave via LDS
  #pragma unroll
  for (int off = 16; off > 0; off >>= 1) {
    #pragma unroll
    for (int w = 0; w < 16; ++w) acc[w] += __shfl_down(acc[w], off, 32);
  }
  int wave = tid >> 5, lane = tid & 31;
  if (lane == 0) {
    #pragma unroll
    for (int w = 0; w < 16; ++w) red[wave * 16 + w] = acc[w];
  }
  __syncthreads();
  if (tid < 12) {   // keep only wires 0..11
    float sum = 0.f;
    #pragma unroll
    for (int wv = 0; wv < 32; ++wv) sum += red[wv * 16 + tid];
    feats[b * 24 + k * 12 + tid] = sum;
  }
}

// ---------------------------------------------------------------------------
// Kernel C: MLP 24->128->64->4 on batch 16, single wave32, f32 WMMA 16x16x4.
// Fragment layouts (ISA 7.12.2):
//   A 16x4 : lanes 0-15 = M, VGPR0/1 = K{0,1}; lanes 16-31 = M, K{2,3}
//   B 4x16 : lanes 0-15 = N, VGPR0/1 = K{0,1}; lanes 16-31 = N, K{2,3}
//   C/D    : lane half = N, VGPR r -> M = r (+8 in high half)
// ---------------------------------------------------------------------------
__global__ void __launch_bounds__(32)
mlp_kernel(const float* __restrict__ feats,
           const float* __restrict__ w1, const float* __restrict__ b1,
           const float* __restrict__ w2, const float* __restrict__ b2,
           const float* __restrict__ w3, const float* __restrict__ b3,
           float* __restrict__ out) {
  __shared__ float A0[16 * 24];
  __shared__ float H1[16 * 128];
  __shared__ float H2[16 * 64];
  const int lane = threadIdx.x;
  const int half = lane >> 4;     // 0: K{0,1}, 1: K{2,3}
  const int l = lane & 15;        // M for A-frag, N for B/C/D-frag

  for (int i = lane; i < 16 * 24; i += 32) A0[i] = feats[i];
  __syncthreads();

  // fc1: (16x24) @ (24x128)^T-weights, relu
  for (int nt = 0; nt < 8; ++nt) {
    v8f acc = {};
    const int n = nt * 16 + l;
    #pragma unroll
    for (int kk = 0; kk < 24; kk += 4) {
      const int kb = kk + half * 2;
      v2f a;  a.x = A0[l * 24 + kb];      a.y = A0[l * 24 + kb + 1];
      v2f bb; bb.x = w1[n * 24 + kb];     bb.y = w1[n * 24 + kb + 1];
      acc = __builtin_amdgcn_wmma_f32_16x16x4_f32(false, a, false, bb,
                                                  (short)0, acc, false, false);
    }
    const float bias = b1[n];
    #pragma unroll
    for (int r = 0; r < 8; ++r) {
      float v = acc[r] + bias;
      H1[(r + half * 8) * 128 + n] = v > 0.f ? v : 0.f;
    }
  }
  __syncthreads();

  // fc2: (16x128) @ (128x64)^T-weights, relu
  for (int nt = 0; nt < 4; ++nt) {
    v8f acc = {};
    const int n = nt * 16 + l;
    for (int kk = 0; kk < 128; kk += 4) {
      const int kb = kk + half * 2;
      v2f a;  a.x = H1[l * 128 + kb];     a.y = H1[l * 128 + kb + 1];
      v2f bb; bb.x = w2[n * 128 + kb];    bb.y = w2[n * 128 + kb + 1];
      acc = __builtin_amdgcn_wmma_f32_16x16x4_f32(false, a, false, bb,
                                                  (short)0, acc, false, false);
    }
    const float bias = b2[n];
    #pragma unroll
    for (int r = 0; r < 8; ++r) {
      float v = acc[r] + bias;
      H2[(r + half * 8) * 64 + n] = v > 0.f ? v : 0.f;
    }
  }
  __syncthreads();

  // fc3: (16x64) @ (64x4)^T-weights, N padded to 16 (zero columns 4..15)
  {
    v8f acc = {};
    const int n = l;
    const bool nv = (n < 4);
    for (int kk = 0; kk < 64; kk += 4) {
      const int kb = kk + half * 2;
      v2f a;  a.x = H2[l * 64 + kb];      a.y = H2[l * 64 + kb + 1];
      v2f bb;
      bb.x = nv ? w3[n * 64 + kb]     : 0.f;
      bb.y = nv ? w3[n * 64 + kb + 1] : 0.f;
      acc = __builtin_amdgcn_wmma_f32_16x16x4_f32(false, a, false, bb,
                                                  (short)0, acc, false, false);
    }
    if (nv) {
      const float bias = b3[n];
      #pragma unroll
      for (int r = 0; r < 8; ++r)
        out[(r + half * 8) * 4 + n] = acc[r] + bias;
    }
  }
}

// ---------------------------------------------------------------------------
// Launch: ws layout (floats): [0..255] xd, [256..639] feats,
// [1024 ..) state real (32*65536), then state imag (32*65536). ~16.8 MB.
// ---------------------------------------------------------------------------
extern "C" void kernel_launch(void* const* d_in, const int* in_sizes, int n_in,
                              void* d_out, int out_size, void* d_ws, size_t ws_size,
                              hipStream_t stream) {
  const float* x  = (const float*)d_in[0];
  const float* kp = (const float*)d_in[1];
  const float* w1 = (const float*)d_in[2];
  const float* b1 = (const float*)d_in[3];
  const float* w2 = (const float*)d_in[4];
  const float* b2 = (const float*)d_in[5];
  const float* w3 = (const float*)d_in[6];
  const float* b3 = (const float*)d_in[7];

  float* ws    = (float*)d_ws;
  float* xd    = ws;                       // 256
  float* feats = ws + 256;                 // 384
  float* re    = ws + 1024;                // 32*65536
  float* im    = re + 32 * (size_t)NST;    // 32*65536

  avgpool_kernel<<<1, 256, 0, stream>>>(x, xd);
  qsim_kernel<<<32, NTH, 0, stream>>>(xd, kp, re, im, feats);
  mlp_kernel<<<1, 32, 0, stream>>>(feats, w1, b1, w2, b2, w3, b3, (float*)d_out);
}